// OrdinalPooling2D_31842887532985
// MI455X (gfx1250) — compile-verified
//
#include <hip/hip_runtime.h>

// OrdinalPooling2D: 2x2/stride-2 pool, per-channel sorted-window weighted sum.
// x: [N,112,112,128] f32 NHWC, w: [128,4] f32  ->  out: [N,56,56,128] f32.
// Pure HBM-streaming problem (~257 MB @ 23.3 TB/s => ~11 us floor).
// Strategy: wave32 lane-group = 4 channels -> all traffic is full 512B b128
// transactions; persistent grid-stride threads hoist per-channel weights into
// registers; NT cache hints keep the one-touch stream from thrashing L2.

#define H_IN 112
#define W_IN 112
#define C_CH 128
#define H_OUT 56
#define W_OUT 56

// Native clang vector type: required by __builtin_nontemporal_load/store
// (HIP's float4 is a struct and is rejected). Lowers to b128 global ops.
typedef float v4f __attribute__((ext_vector_type(4)));

// ---------------------------------------------------------------------------
// Kernel 1: clip negatives, renormalize per channel, transpose to SoA planes
// wk[k*128 + c] so the main kernel reads 4 coalesced float4s from a 2KB table.
// ---------------------------------------------------------------------------
__global__ void ordpool_prep_weights(const float* __restrict__ w,
                                     float* __restrict__ wk) {
    const int c = threadIdx.x;            // 0..127, one thread per channel
    float w0 = w[c * 4 + 0];
    float w1 = w[c * 4 + 1];
    float w2 = w[c * 4 + 2];
    float w3 = w[c * 4 + 3];
    w0 = (w0 >= 0.0f) ? w0 : 0.0f;
    w1 = (w1 >= 0.0f) ? w1 : 0.0f;
    w2 = (w2 >= 0.0f) ? w2 : 0.0f;
    w3 = (w3 >= 0.0f) ? w3 : 0.0f;
    const float s = w0 + w1 + w2 + w3;
    wk[0 * C_CH + c] = w0 / s;
    wk[1 * C_CH + c] = w1 / s;
    wk[2 * C_CH + c] = w2 / s;
    wk[3 * C_CH + c] = w3 / s;
}

// ---------------------------------------------------------------------------
// Kernel 2: streaming ordinal pool, persistent grid-stride.
// One work item = one output pixel x 4 channels; a wave32 covers one pixel's
// full 128 channels. Grid stride is a multiple of 32, so each thread's
// channel group (and its 16 weight values) is loop-invariant.
// ---------------------------------------------------------------------------
__device__ __forceinline__ void cswap_desc(float& hi, float& lo) {
    const float mx = fmaxf(hi, lo);
    const float mn = fminf(hi, lo);
    hi = mx;
    lo = mn;
}

__global__ void __launch_bounds__(256)
ordpool_main(const float* __restrict__ x,
             const float* __restrict__ wk,
             float* __restrict__ out,
             int total) {
    const int tid0   = blockIdx.x * blockDim.x + threadIdx.x;
    const int stride = gridDim.x * blockDim.x;   // multiple of 32

    const int c4   = tid0 & 31;                  // loop-invariant channel group
    const int cOff = c4 * 4;
    const size_t rowStride = (size_t)W_IN * C_CH;

    // Per-channel ordinal weights: hoisted once into registers (2KB table,
    // temporal policy -> stays L2/L0 resident).
    const v4f wk0 = *(const v4f*)(wk + 0 * C_CH + cOff);
    const v4f wk1 = *(const v4f*)(wk + 1 * C_CH + cOff);
    const v4f wk2 = *(const v4f*)(wk + 2 * C_CH + cOff);
    const v4f wk3 = *(const v4f*)(wk + 3 * C_CH + cOff);

    for (int idx = tid0; idx < total; idx += stride) {
        const int pix = idx >> 5;                // output pixel
        const int wo  = pix % W_OUT;
        const int t   = pix / W_OUT;
        const int ho  = t % H_OUT;
        const int n   = t / H_OUT;

        const size_t base =
            (((size_t)n * H_IN + 2 * ho) * W_IN + 2 * wo) * C_CH + cOff;

        // Window corners in (i over KH, j over KW) reference order.
        // One-touch stream: non-temporal loads (TH=NT) so x doesn't churn L2.
        const v4f p0 =
            __builtin_nontemporal_load((const v4f*)(x + base));
        const v4f p1 =
            __builtin_nontemporal_load((const v4f*)(x + base + C_CH));
        const v4f p2 =
            __builtin_nontemporal_load((const v4f*)(x + base + rowStride));
        const v4f p3 =
            __builtin_nontemporal_load((const v4f*)(x + base + rowStride + C_CH));

        // Warm L2 one output row ahead (global_prefetch_b8, speculative).
        if (ho + 1 < H_OUT) {
            __builtin_prefetch(x + base + 2 * rowStride, 0, 0);
        }

        v4f r;
        // Odd-even transposition, descending: (0,1)(2,3) (1,2) (0,1)(2,3) (1,2)
#define ORDPOOL_SORT_DOT(comp)                                                 \
        {                                                                      \
            float s0 = p0.comp, s1 = p1.comp, s2 = p2.comp, s3 = p3.comp;      \
            cswap_desc(s0, s1);                                                \
            cswap_desc(s2, s3);                                                \
            cswap_desc(s1, s2);                                                \
            cswap_desc(s0, s1);                                                \
            cswap_desc(s2, s3);                                                \
            cswap_desc(s1, s2);                                                \
            r.comp = fmaf(s0, wk0.comp,                                        \
                     fmaf(s1, wk1.comp,                                        \
                     fmaf(s2, wk2.comp, s3 * wk3.comp)));                      \
        }
        ORDPOOL_SORT_DOT(x)
        ORDPOOL_SORT_DOT(y)
        ORDPOOL_SORT_DOT(z)
        ORDPOOL_SORT_DOT(w)
#undef ORDPOOL_SORT_DOT

        const size_t obase =
            (((size_t)n * H_OUT + ho) * W_OUT + wo) * C_CH + cOff;
        // Written once, never re-read by this dispatch -> NT store.
        __builtin_nontemporal_store(r, (v4f*)(out + obase));
    }
}

// ---------------------------------------------------------------------------
extern "C" void kernel_launch(void* const* d_in, const int* in_sizes, int n_in,
                              void* d_out, int out_size, void* d_ws, size_t ws_size,
                              hipStream_t stream) {
    const float* x = (const float*)d_in[0];             // [N,112,112,128]
    const float* w = (const float*)d_in[1];             // [128,4]
    float* out = (float*)d_out;                         // [N,56,56,128]
    float* wk  = (float*)d_ws;                          // 512 floats (2KB) SoA weights

    const int N = in_sizes[0] / (H_IN * W_IN * C_CH);

    ordpool_prep_weights<<<1, C_CH, 0, stream>>>(w, wk);

    const int total = N * H_OUT * W_OUT * (C_CH / 4);   // one item per 4 channels
    const int block = 256;
    // Persistent grid: enough waves to saturate HBM, ~3 pixels per thread.
    int grid = (total + block - 1) / block;
    if (grid > 4096) grid = 4096;
    ordpool_main<<<grid, block, 0, stream>>>(x, wk, out, total);
}